// NoisyTopExpertsPerItemRouter_16157666967664
// MI455X (gfx1250) — compile-verified
//
#include <hip/hip_runtime.h>
#include <math.h>

// ---- CDNA5 WMMA fragment types -------------------------------------------
typedef __attribute__((ext_vector_type(2))) float v2f;  // A/B frag: 16x4 / 4x16 f32
typedef __attribute__((ext_vector_type(4))) float v4f;
typedef __attribute__((ext_vector_type(8))) float v8f;  // C/D frag: 16x16 f32
typedef __attribute__((ext_vector_type(4))) int   v4i_t;

#define NROWS 8192
#define DIM   4096
#define NEXP  64
#define EPS   1e-6f

// out layout: [0,16384) combine_weights | [16384,32768) indices | [32768] aux
#define IDX_BASE  (2 * NROWS)
#define AUX_BASE  (4 * NROWS)

// ---- split-K GEMM config --------------------------------------------------
#define KSPLIT   4
#define KSEG     (DIM / KSPLIT)          // 1024 k per block
#define CK       64                      // k-chunk staged in LDS
#define NCHUNK   (KSEG / CK)             // 16
#define WSTRIDE  (CK + 4)                // 68 floats: 16B-aligned rows, conflict-free
#define ATHREADS 128                     // 4 waves
#define AROWS    64                      // rows per block (16 per wave)

#define BTHREADS 256
#define NBLOCKS_B (NROWS / BTHREADS)     // 32

// ---- async global->LDS path (guarded) ------------------------------------
#if defined(__has_builtin)
#if __has_builtin(__builtin_amdgcn_global_load_async_to_lds_b128) && \
    __has_builtin(__builtin_amdgcn_s_wait_asynccnt)
#define HAS_ASYNC_LDS 1
#endif
#endif
#ifndef HAS_ASYNC_LDS
#define HAS_ASYNC_LDS 0
#endif

__device__ __forceinline__ void copy16_to_lds(float* dst_lds, const float* src) {
#if HAS_ASYNC_LDS
    // signature (from clang diagnostic): (v4i AS1*, v4i AS3*, imm offset, imm cpol)
    __builtin_amdgcn_global_load_async_to_lds_b128(
        (__attribute__((address_space(1))) v4i_t*)src,
        (__attribute__((address_space(3))) v4i_t*)dst_lds,
        0, 0);
#else
    *(v4f*)dst_lds = *(const v4f*)src;
#endif
}

#if HAS_ASYNC_LDS
#define WAIT_ASYNC(n) __builtin_amdgcn_s_wait_asynccnt(n)
#else
#define WAIT_ASYNC(n) ((void)0)
#endif

// ===========================================================================
// Kernel A: split-K partial GEMM.  grid (128 row-blocks, 4 k-splits).
// Each wave: 16 rows x 64 experts over 1024 k, W chunk staged in LDS
// (double-buffered async b128), A streamed from HBM, V_WMMA_F32_16X16X4_F32.
// ===========================================================================
__global__ void __launch_bounds__(ATHREADS)
router_gemm_part(const float* __restrict__ x, const float* __restrict__ W,
                 float* __restrict__ part)
{
    __shared__ float lds_w[2][NEXP * WSTRIDE];   // 2 x 17KB

    const int tid     = threadIdx.x;
    const int wave    = tid >> 5;
    const int lane    = tid & 31;
    const int lane_lo = lane & 15;
    const int lane_hi = lane >> 4;

    const int rb = blockIdx.x;           // row block
    const int ks = blockIdx.y;           // k split
    const int k0 = ks * KSEG;
    const int row_base = rb * AROWS + wave * 16;

    const float* wsrc = W + k0;          // + e*DIM + chunk*CK + 4*g

    // issue one chunk (64 experts x 64 k = 16KB) as 1024 b128 copies, 8/thread
    auto issue = [&](int c, int b) {
        const float* src = wsrc + c * CK;
        float* dst = &lds_w[b][0];
        #pragma unroll
        for (int i = 0; i < 8; ++i) {
            const int idx = tid + ATHREADS * i;      // [0,1024)
            const int e = idx >> 4;
            const int g = idx & 15;
            copy16_to_lds(dst + e * WSTRIDE + 4 * g,
                          src + (size_t)e * DIM + 4 * g);
        }
    };

    issue(0, 0);
    issue(1, 1);

    const float* ap = x + (size_t)(row_base + lane_lo) * DIM + k0 + 2 * lane_hi;

    v8f acc0 = {}, acc1 = {}, acc2 = {}, acc3 = {};
    for (int c = 0; c < NCHUNK; ++c) {
        const int b = c & 1;
        WAIT_ASYNC(8);          // chunk c landed (in-order); c+1 may still fly
        __syncthreads();

        const float* bw = &lds_w[b][lane_lo * WSTRIDE + 2 * lane_hi];
        const float* ac = ap + c * CK;
        #pragma unroll
        for (int kk = 0; kk < CK; kk += 4) {
            v2f a  = *(const v2f*)(ac + kk);
            v2f f0 = *(const v2f*)(bw + kk);
            v2f f1 = *(const v2f*)(bw + 16 * WSTRIDE + kk);
            v2f f2 = *(const v2f*)(bw + 32 * WSTRIDE + kk);
            v2f f3 = *(const v2f*)(bw + 48 * WSTRIDE + kk);
            acc0 = __builtin_amdgcn_wmma_f32_16x16x4_f32(false, a, false, f0, (short)0, acc0, false, false);
            acc1 = __builtin_amdgcn_wmma_f32_16x16x4_f32(false, a, false, f1, (short)0, acc1, false, false);
            acc2 = __builtin_amdgcn_wmma_f32_16x16x4_f32(false, a, false, f2, (short)0, acc2, false, false);
            acc3 = __builtin_amdgcn_wmma_f32_16x16x4_f32(false, a, false, f3, (short)0, acc3, false, false);
        }
        __syncthreads();        // everyone done reading buf b
        if (c + 2 < NCHUNK) issue(c + 2, b);
    }

    // D layout: VGPR v, lane L -> D[M = v + 8*(L>>4)][N = L&15]
    float* pr = part + ((size_t)ks * NROWS + row_base + 8 * lane_hi) * NEXP + lane_lo;
    #pragma unroll
    for (int v = 0; v < 8; ++v) {
        pr[(size_t)v * NEXP +  0] = acc0[v];
        pr[(size_t)v * NEXP + 16] = acc1[v];
        pr[(size_t)v * NEXP + 32] = acc2[v];
        pr[(size_t)v * NEXP + 48] = acc3[v];
    }
}

// ===========================================================================
// Kernel B: sum K-partials, softmax, top-2 + renorm, deterministic importance
// ===========================================================================
__global__ void __launch_bounds__(BTHREADS)
router_softmax_topk(const float* __restrict__ part, float* __restrict__ out,
                    float* __restrict__ blockimp)
{
    __shared__ float wimp[BTHREADS / 32][NEXP];

    const int tid  = threadIdx.x;
    const int wave = tid >> 5;
    const int lane = tid & 31;
    const int r    = blockIdx.x * BTHREADS + tid;

    float l[NEXP];
    {
        const float* p = part + (size_t)r * NEXP;
        #pragma unroll
        for (int e = 0; e < NEXP; ++e) l[e] = p[e];
        #pragma unroll
        for (int ksp = 1; ksp < KSPLIT; ++ksp) {
            const float* q = p + (size_t)ksp * NROWS * NEXP;
            #pragma unroll
            for (int e = 0; e < NEXP; ++e) l[e] += q[e];
        }
    }

    float mx = l[0];
    #pragma unroll
    for (int e = 1; e < NEXP; ++e) mx = fmaxf(mx, l[e]);

    float sum = 0.0f;
    float b1v = -1e30f, b2v = -1e30f;
    int   b1i = 0,      b2i = 0;
    #pragma unroll
    for (int e = 0; e < NEXP; ++e) {
        float g = expf(l[e] - mx);
        l[e] = g;
        sum += g;
        if (g > b1v)      { b2v = b1v; b2i = b1i; b1v = g; b1i = e; }
        else if (g > b2v) { b2v = g;   b2i = e; }
    }
    const float inv = 1.0f / sum;

    const float g1 = b1v * inv, g2 = b2v * inv;
    const float c1 = 1.0f / (1.0f + expf(g2 - g1));
    out[2 * r + 0]            = c1;
    out[2 * r + 1]            = 1.0f - c1;
    out[IDX_BASE + 2 * r + 0] = (float)b1i;
    out[IDX_BASE + 2 * r + 1] = (float)b2i;

    // deterministic importance: butterfly over the wave, then fixed-order sums
    #pragma unroll
    for (int e = 0; e < NEXP; ++e) {
        float g = l[e] * inv;
        g += __shfl_xor(g, 16);
        g += __shfl_xor(g, 8);
        g += __shfl_xor(g, 4);
        g += __shfl_xor(g, 2);
        g += __shfl_xor(g, 1);
        if (lane == 0) wimp[wave][e] = g;
    }
    __syncthreads();
    if (tid < NEXP) {
        float s = 0.0f;
        #pragma unroll
        for (int w = 0; w < BTHREADS / 32; ++w) s += wimp[w][tid];
        blockimp[blockIdx.x * NEXP + tid] = s;
    }
}

__global__ void __launch_bounds__(NEXP)
router_finalize_split(const float* __restrict__ blockimp, float* __restrict__ out)
{
    __shared__ float s[NEXP];
    const int t = threadIdx.x;
    float v = 0.0f;
    #pragma unroll
    for (int b = 0; b < NBLOCKS_B; ++b) v += blockimp[b * NEXP + t];
    s[t] = v;
    __syncthreads();
    if (t == 0) {
        float m = 0.0f;
        #pragma unroll
        for (int e = 0; e < NEXP; ++e) m += s[e];
        m *= (1.0f / NEXP);
        float ss = 0.0f;
        #pragma unroll
        for (int e = 0; e < NEXP; ++e) { float d = s[e] - m; ss += d * d; }
        const float stdv = sqrtf(ss / (float)(NEXP - 1));
        const float cv   = stdv / (m + EPS);
        out[AUX_BASE] = cv * cv;
    }
}

// ===========================================================================
// Fallback (small ws): round-1 monolithic kernel
// ===========================================================================
#define LSTRIDE 65
__global__ void __launch_bounds__(NEXP)
router_zero_ws(float* __restrict__ gimp) { gimp[threadIdx.x] = 0.0f; }

__global__ void __launch_bounds__(ATHREADS)
router_mono(const float* __restrict__ x, const float* __restrict__ W,
            float* __restrict__ out, float* __restrict__ gimp)
{
    __shared__ float lds_logits[AROWS * LSTRIDE];
    __shared__ float lds_imp[NEXP];

    const int tid = threadIdx.x, wave = tid >> 5, lane = tid & 31;
    const int lane_lo = lane & 15, lane_hi = lane >> 4;
    if (tid < NEXP) lds_imp[tid] = 0.0f;

    const int row_base = blockIdx.x * AROWS + wave * 16;
    const float* aptr = x + (size_t)(row_base + lane_lo) * DIM + 2 * lane_hi;
    const float* bp0  = W + (size_t)(lane_lo +  0) * DIM + 2 * lane_hi;
    const float* bp1  = W + (size_t)(lane_lo + 16) * DIM + 2 * lane_hi;
    const float* bp2  = W + (size_t)(lane_lo + 32) * DIM + 2 * lane_hi;
    const float* bp3  = W + (size_t)(lane_lo + 48) * DIM + 2 * lane_hi;

    v8f acc0 = {}, acc1 = {}, acc2 = {}, acc3 = {};
    #pragma unroll 4
    for (int k = 0; k < DIM; k += 4) {
        v2f a  = *(const v2f*)(aptr + k);
        v2f f0 = *(const v2f*)(bp0 + k);
        v2f f1 = *(const v2f*)(bp1 + k);
        v2f f2 = *(const v2f*)(bp2 + k);
        v2f f3 = *(const v2f*)(bp3 + k);
        acc0 = __builtin_amdgcn_wmma_f32_16x16x4_f32(false, a, false, f0, (short)0, acc0, false, false);
        acc1 = __builtin_amdgcn_wmma_f32_16x16x4_f32(false, a, false, f1, (short)0, acc1, false, false);
        acc2 = __builtin_amdgcn_wmma_f32_16x16x4_f32(false, a, false, f2, (short)0, acc2, false, false);
        acc3 = __builtin_amdgcn_wmma_f32_16x16x4_f32(false, a, false, f3, (short)0, acc3, false, false);
    }
    {
        float* base = lds_logits + (wave * 16 + 8 * lane_hi) * LSTRIDE + lane_lo;
        #pragma unroll
        for (int v = 0; v < 8; ++v) {
            base[v * LSTRIDE +  0] = acc0[v];
            base[v * LSTRIDE + 16] = acc1[v];
            base[v * LSTRIDE + 32] = acc2[v];
            base[v * LSTRIDE + 48] = acc3[v];
        }
    }
    __syncthreads();

    if (tid < AROWS) {
        float* lrow = lds_logits + tid * LSTRIDE;
        float mx = lrow[0];
        #pragma unroll
        for (int e = 1; e < NEXP; ++e) mx = fmaxf(mx, lrow[e]);
        float sum = 0.0f, b1v = -1e30f, b2v = -1e30f;
        int b1i = 0, b2i = 0;
        #pragma unroll
        for (int e = 0; e < NEXP; ++e) {
            float g = expf(lrow[e] - mx);
            lrow[e] = g; sum += g;
            if (g > b1v)      { b2v = b1v; b2i = b1i; b1v = g; b1i = e; }
            else if (g > b2v) { b2v = g;   b2i = e; }
        }
        const float inv = 1.0f / sum;
        #pragma unroll
        for (int e = 0; e < NEXP; ++e) atomicAdd(&lds_imp[e], lrow[e] * inv);
        const float g1 = b1v * inv, g2 = b2v * inv;
        const float c1 = 1.0f / (1.0f + expf(g2 - g1));
        const int r = blockIdx.x * AROWS + tid;
        out[2 * r + 0] = c1;
        out[2 * r + 1] = 1.0f - c1;
        out[IDX_BASE + 2 * r + 0] = (float)b1i;
        out[IDX_BASE + 2 * r + 1] = (float)b2i;
    }
    __syncthreads();
    if (tid < NEXP) atomicAdd(&gimp[tid], lds_imp[tid]);
}

__global__ void __launch_bounds__(NEXP)
router_finalize(const float* __restrict__ gimp, float* __restrict__ out)
{
    __shared__ float s[NEXP];
    const int t = threadIdx.x;
    s[t] = gimp[t];
    __syncthreads();
    if (t == 0) {
        float m = 0.0f;
        #pragma unroll
        for (int e = 0; e < NEXP; ++e) m += s[e];
        m *= (1.0f / NEXP);
        float ss = 0.0f;
        #pragma unroll
        for (int e = 0; e < NEXP; ++e) { float d = s[e] - m; ss += d * d; }
        const float stdv = sqrtf(ss / (float)(NEXP - 1));
        const float cv   = stdv / (m + EPS);
        out[AUX_BASE] = cv * cv;
    }
}

// ===========================================================================
extern "C" void kernel_launch(void* const* d_in, const int* in_sizes, int n_in,
                              void* d_out, int out_size, void* d_ws, size_t ws_size,
                              hipStream_t stream)
{
    const float* x = (const float*)d_in[0];   // [8192, 4096]
    const float* W = (const float*)d_in[1];   // [64, 4096]
    float* out = (float*)d_out;

    const size_t part_elems = (size_t)KSPLIT * NROWS * NEXP;          // 2M floats
    const size_t need = (part_elems + (size_t)NBLOCKS_B * NEXP) * sizeof(float);

    if (ws_size >= need) {
        float* part     = (float*)d_ws;
        float* blockimp = part + part_elems;
        dim3 grid(NROWS / AROWS, KSPLIT);                              // (128, 4)
        router_gemm_part<<<grid, ATHREADS, 0, stream>>>(x, W, part);
        router_softmax_topk<<<NBLOCKS_B, BTHREADS, 0, stream>>>(part, out, blockimp);
        router_finalize_split<<<1, NEXP, 0, stream>>>(blockimp, out);
    } else {
        float* gimp = (float*)d_ws;
        router_zero_ws<<<1, NEXP, 0, stream>>>(gimp);
        router_mono<<<NROWS / AROWS, ATHREADS, 0, stream>>>(x, W, out, gimp);
        router_finalize<<<1, NEXP, 0, stream>>>(gimp, out);
    }
}